// StandGCN1_22428319219736
// MI455X (gfx1250) — compile-verified
//
#include <hip/hip_runtime.h>

typedef float v2f __attribute__((ext_vector_type(2)));
typedef float v8f __attribute__((ext_vector_type(8)));

// ---------------------------------------------------------------------------
// Pass 1: zero the accumulator (d_out) and set deg = 1.0 (self-loop).
// ---------------------------------------------------------------------------
__global__ void gcn_init_kernel(float* __restrict__ out, float* __restrict__ deg,
                                int total, int n) {
    int i = blockIdx.x * blockDim.x + threadIdx.x;
    if (i < total) out[i] = 0.0f;
    if (i < n) deg[i] = 1.0f;
}

// ---------------------------------------------------------------------------
// Pass 2: degree of each dst node (float atomics; values are small counts).
// ---------------------------------------------------------------------------
__global__ void gcn_degree_kernel(const int* __restrict__ dst,
                                  float* __restrict__ deg, int E) {
    int e = blockIdx.x * blockDim.x + threadIdx.x;
    if (e < E) atomicAdd(&deg[dst[e]], 1.0f);
}

// ---------------------------------------------------------------------------
// Pass 3: dinv = rsqrt(deg)
// ---------------------------------------------------------------------------
__global__ void gcn_rsqrt_kernel(const float* __restrict__ deg,
                                 float* __restrict__ dinv, int n) {
    int i = blockIdx.x * blockDim.x + threadIdx.x;
    if (i < n) dinv[i] = rsqrtf(deg[i]);
}

// ---------------------------------------------------------------------------
// Pass 4: h = x @ W via V_WMMA_F32_16X16X4_F32.
// One wave computes one 16x16 tile of h. Block = 4 waves covering F_OUT=64.
// VGPR layouts per CDNA5 ISA 7.12.2:
//   A (16x4 f32):  lanes 0-15 -> K = k0+{0,1}; lanes 16-31 -> K = k0+{2,3}
//   B (4x16 f32):  vgpr0 -> K = k0 + 2*half ; vgpr1 -> K = k0 + 2*half + 1
//   C/D (16x16):   vgpr i -> row M = i + 8*half, col N = lane%16
// ---------------------------------------------------------------------------
__global__ void gcn_gemm_wmma_kernel(const float* __restrict__ x,
                                     const float* __restrict__ W,
                                     float* __restrict__ h,
                                     int N, int FIN, int FOUT) {
    const int lane = threadIdx.x & 31;
    const int wave = threadIdx.x >> 5;
    const int half = lane >> 4;      // 0: lanes 0-15, 1: lanes 16-31
    const int l    = lane & 15;

    const int m0 = blockIdx.x << 4;  // output row tile
    const int n0 = wave << 4;        // output col tile

    int arow = m0 + l;
    if (arow >= N) arow = N - 1;     // clamp (EXEC must stay all-ones for WMMA)

    const float* __restrict__ ap = x + (size_t)arow * FIN + 2 * half;
    const float* __restrict__ bp = W + (size_t)(2 * half) * FOUT + n0 + l;

    v8f c = {};
    for (int k0 = 0; k0 < FIN; k0 += 4) {
        v2f a;
        a.x = ap[k0];
        a.y = ap[k0 + 1];            // contiguous -> global_load_b64
        v2f b;
        b.x = bp[(size_t)k0 * FOUT];
        b.y = bp[(size_t)(k0 + 1) * FOUT];
        // 8 args: (neg_a, A, neg_b, B, c_mod, C, reuse_a, reuse_b)
        c = __builtin_amdgcn_wmma_f32_16x16x4_f32(
                false, a, false, b, (short)0, c, false, false);
    }

#pragma unroll
    for (int i = 0; i < 8; ++i) {
        int row = m0 + i + half * 8;
        if (row < N) h[(size_t)row * FOUT + n0 + l] = c[i];
    }
}

// ---------------------------------------------------------------------------
// Pass 5: edge scatter. One wave32 per edge, 2 features per lane.
// Gather of h[src] is a coalesced 256B read per wave; src/dst/dinv loads are
// wave-uniform. Accumulation via global_atomic_add_f32 (no return).
// ---------------------------------------------------------------------------
__global__ void gcn_scatter_kernel(const int* __restrict__ src,
                                   const int* __restrict__ dst,
                                   const float* __restrict__ h,
                                   const float* __restrict__ dinv,
                                   float* __restrict__ acc,
                                   long long E, int FOUT) {
    long long t = (long long)blockIdx.x * blockDim.x + threadIdx.x;
    long long e = t >> 5;
    if (e >= E) return;
    const int lane = (int)(t & 31);

    const int s = src[e];
    const int d = dst[e];
    const float w = dinv[s] * dinv[d];

    const float* __restrict__ hp = h + (size_t)s * FOUT;
    float* __restrict__ op = acc + (size_t)d * FOUT;
    for (int f = lane * 2; f < FOUT; f += 64) {
        atomicAdd(&op[f],     w * hp[f]);
        atomicAdd(&op[f + 1], w * hp[f + 1]);
    }
}

// ---------------------------------------------------------------------------
// Pass 6: add self-loop term + bias, apply ReLU.  out = relu(acc + h*dinv^2 + b)
// ---------------------------------------------------------------------------
__global__ void gcn_finalize_kernel(float* __restrict__ out,
                                    const float* __restrict__ h,
                                    const float* __restrict__ dinv,
                                    const float* __restrict__ bias,
                                    int N, int FOUT) {
    int i = blockIdx.x * blockDim.x + threadIdx.x;
    if (i >= N * FOUT) return;
    const int n = i / FOUT;
    const int f = i - n * FOUT;
    const float di = dinv[n];
    float v = out[i] + h[i] * di * di + bias[f];
    out[i] = v > 0.0f ? v : 0.0f;
}

// ---------------------------------------------------------------------------
// kernel_launch
//   d_in[0] = x   [N, F_IN]   f32
//   d_in[1] = adj [2, E]      i32  (row 0 = src, row 1 = dst)
//   d_in[2] = W   [F_IN, F_OUT] f32
//   d_in[3] = b   [F_OUT]     f32
//   d_out   = out [N, F_OUT]  f32
// ---------------------------------------------------------------------------
extern "C" void kernel_launch(void* const* d_in, const int* in_sizes, int n_in,
                              void* d_out, int out_size, void* d_ws, size_t ws_size,
                              hipStream_t stream) {
    const float* x    = (const float*)d_in[0];
    const int*   adj  = (const int*)d_in[1];
    const float* W    = (const float*)d_in[2];
    const float* bias = (const float*)d_in[3];

    const int  FOUT = in_sizes[3];                 // 64
    const int  FIN  = in_sizes[2] / FOUT;          // 256
    const int  N    = in_sizes[0] / FIN;           // 50000
    const long long E = (long long)in_sizes[1] / 2; // 800000

    const int* src = adj;
    const int* dst = adj + E;

    float* out  = (float*)d_out;
    float* h    = (float*)d_ws;                    // N*FOUT floats (12.8 MB)
    float* deg  = h + (size_t)N * FOUT;            // N floats
    float* dinv = deg + N;                         // N floats

    const int total = N * FOUT;

    gcn_init_kernel<<<(total + 255) / 256, 256, 0, stream>>>(out, deg, total, N);

    gcn_degree_kernel<<<((int)E + 255) / 256, 256, 0, stream>>>(dst, deg, (int)E);

    gcn_rsqrt_kernel<<<(N + 255) / 256, 256, 0, stream>>>(deg, dinv, N);

    const int nwaves = FOUT / 16;                  // 4 waves -> 128 threads/block
    gcn_gemm_wmma_kernel<<<(N + 15) / 16, 32 * nwaves, 0, stream>>>(
        x, W, h, N, FIN, FOUT);

    const long long sthreads = E * 32;
    gcn_scatter_kernel<<<(unsigned int)((sthreads + 255) / 256), 256, 0, stream>>>(
        src, dst, h, dinv, out, E, FOUT);

    gcn_finalize_kernel<<<(total + 255) / 256, 256, 0, stream>>>(
        out, h, dinv, bias, N, FOUT);
}